// UnbalancedLPModel_85856396248061
// MI455X (gfx1250) — compile-verified
//
#include <hip/hip_runtime.h>
#include <math.h>

typedef __attribute__((ext_vector_type(2))) float v2f;
typedef __attribute__((ext_vector_type(8))) float v8f;
typedef __attribute__((ext_vector_type(4))) int   v4i;

#define EPSF 1e-15f

// ---------------------------------------------------------------------------
// K1: per-node normalize + fold weights into scalar s = xn.x*w0 - xn.y*w1,
//     computed 16 nodes/wave via V_WMMA_F32_16X16X4_F32 (exact f32 math).
//     Also zeroes the accumulator y[].
// A (16x4): lanes 0-15 hold node (base+lane) features at K=0,1; lanes 16-31
// hold K=2,3 which we set to 0, so B's rows 2,3 are don't-care (layout-robust).
// B (4x16): row0 = w0 for all columns, row1 = -w1 for all columns.
// D (16x16): every column identical; lane l holds rows {r + 8*(l>=16)}.
// ---------------------------------------------------------------------------
__global__ __launch_bounds__(256) void node_prep_kernel(
    const float* __restrict__ x,   // N x 2
    const float* __restrict__ W,   // 2x2 row-major: [[w0,-w0],[-w1,w1]]
    float* __restrict__ s,         // N
    float* __restrict__ y,         // N (zeroed here)
    int N)
{
    int tid  = blockIdx.x * blockDim.x + threadIdx.x;
    int lane = threadIdx.x & 31;
    int wave = tid >> 5;
    int base = wave * 16;
    if (base >= N) return;                 // wave-uniform exit

    float w0  = W[0];
    float w1n = -W[3];                     // -w1

    v2f a; a.x = 0.0f; a.y = 0.0f;
    if (lane < 16) {
        float xv  = x[(base + lane) * 2 + 0];
        float yv  = x[(base + lane) * 2 + 1];
        float nrm = sqrtf(xv * xv + yv * yv) + EPSF;
        a.x = xv / nrm;
        a.y = yv / nrm;
    }
    // Robust under either K-pairing of B's VGPRs (rows 2/3 multiply zero A cols)
    v2f b;
    b.x = (lane < 16) ? w0  : w1n;
    b.y = (lane < 16) ? w1n : 0.0f;

    v8f c = {};
    // (neg_a, A, neg_b, B, c_mod, C, reuse_a, reuse_b)
    c = __builtin_amdgcn_wmma_f32_16x16x4_f32(
            false, a, false, b, (short)0, c, false, false);

    // Extract: lanes 0-7 write nodes base+0..7 (their d[r] is row r);
    //          lanes 16-23 write nodes base+8..15 (their d[r] is row 8+r).
    int grp = lane >> 4;
    int r   = lane & 15;
    if (r < 8) {
        float val = 0.0f;
        #pragma unroll
        for (int k = 0; k < 8; ++k) val = (r == k) ? c[k] : val;
        int node = base + grp * 8 + r;
        s[node] = val;
        y[node] = 0.0f;
    }
}

// ---------------------------------------------------------------------------
// K2/K4: edge scatter: y[dst] += s[src], 4 edges per thread.
// Edge indices are int64 < 2^31: load pairs as b128 (ext-vector v4i), use low
// dwords (.x, .z). NT hints keep the 512 MB/pass index stream from evicting
// the L2-resident 4 MB s[]/y[] tables.
// ---------------------------------------------------------------------------
__global__ __launch_bounds__(256) void edge_scatter_kernel(
    const v4i* __restrict__ src_pairs,   // E/2 entries, 2 edges each
    const v4i* __restrict__ dst_pairs,
    const float* __restrict__ s,
    float* __restrict__ y,
    long long npairs)
{
    long long t = (long long)blockIdx.x * blockDim.x + threadIdx.x;
    long long p = t * 2;
    if (p >= npairs) return;

    v4i sp0 = __builtin_nontemporal_load(&src_pairs[p]);
    v4i dp0 = __builtin_nontemporal_load(&dst_pairs[p]);
    bool second = (p + 1) < npairs;
    v4i sp1 = sp0, dp1 = dp0;
    if (second) {
        sp1 = __builtin_nontemporal_load(&src_pairs[p + 1]);
        dp1 = __builtin_nontemporal_load(&dst_pairs[p + 1]);
    }
    float v0 = s[sp0.x];
    float v1 = s[sp0.z];
    atomicAdd(&y[dp0.x], v0);
    atomicAdd(&y[dp0.z], v1);
    if (second) {
        float v2 = s[sp1.x];
        float v3 = s[sp1.z];
        atomicAdd(&y[dp1.x], v2);
        atomicAdd(&y[dp1.z], v3);
    }
}

// Scalar fallback (only used if E is odd -> b128 alignment breaks)
__global__ __launch_bounds__(256) void edge_scatter_scalar_kernel(
    const long long* __restrict__ src,
    const long long* __restrict__ dst,
    const float* __restrict__ s,
    float* __restrict__ y,
    long long E)
{
    long long i = (long long)blockIdx.x * blockDim.x + threadIdx.x;
    if (i >= E) return;
    int si = (int)src[i];
    int di = (int)dst[i];
    atomicAdd(&y[di], s[si]);
}

// ---------------------------------------------------------------------------
// K3: mid transform. q = y[i]; normalized vec = (relu(q),relu(-q))/(|q|+eps);
// fold weights: s[i] = vx*w0 - vy*w1. Re-zero y[i] for pass 2.
// ---------------------------------------------------------------------------
__global__ __launch_bounds__(256) void node_mid_kernel(
    const float* __restrict__ W,
    float* __restrict__ s,
    float* __restrict__ y,
    int N)
{
    int i = blockIdx.x * blockDim.x + threadIdx.x;
    if (i >= N) return;
    float q   = y[i];
    float inv = 1.0f / (fabsf(q) + EPSF);   // norm of (relu(q),relu(-q)) == |q|
    float vx  = fmaxf(q, 0.0f) * inv;
    float vy  = fmaxf(-q, 0.0f) * inv;
    s[i] = vx * W[0] - vy * W[3];
    y[i] = 0.0f;
}

// ---------------------------------------------------------------------------
// K5: epilogue. x2 = (relu(q), relu(-q)); out = sigmoid(x2 . final_weight).
// ---------------------------------------------------------------------------
__global__ __launch_bounds__(256) void node_out_kernel(
    const float* __restrict__ fw,
    const float* __restrict__ y,
    float* __restrict__ out,
    int N)
{
    int i = blockIdx.x * blockDim.x + threadIdx.x;
    if (i >= N) return;
    float q = y[i];
    float z = fmaxf(q, 0.0f) * fw[0] + fmaxf(-q, 0.0f) * fw[1];
    out[i] = 1.0f / (1.0f + expf(-z));
}

extern "C" void kernel_launch(void* const* d_in, const int* in_sizes, int n_in,
                              void* d_out, int out_size, void* d_ws, size_t ws_size,
                              hipStream_t stream) {
    (void)n_in; (void)out_size; (void)ws_size;
    const float*     x  = (const float*)d_in[0];        // (N,2) f32
    const long long* ei = (const long long*)d_in[1];    // (2,E) i64: row0=src, row1=dst
    const float*     W  = (const float*)d_in[2];        // (2,2) f32
    const float*     fw = (const float*)d_in[3];        // (2,)  f32
    float*           out = (float*)d_out;

    const int       N = in_sizes[0] / 2;
    const long long E = (long long)in_sizes[1] / 2;

    float* s = (float*)d_ws;      // N floats
    float* y = s + N;             // N floats (accumulator)

    const long long src_off = 0;
    const long long dst_off = E;

    // K1: normalize + weight-fold via WMMA; zero y
    {
        long long waves   = (N + 15) / 16;
        long long threads = waves * 32;
        int blocks = (int)((threads + 255) / 256);
        node_prep_kernel<<<blocks, 256, 0, stream>>>(x, W, s, y, N);
    }

    // Two identical edge scatter passes with a mid transform between them
    for (int pass = 0; pass < 2; ++pass) {
        if ((E & 1LL) == 0) {
            long long npairs  = E / 2;
            long long threads = (npairs + 1) / 2;
            int blocks = (int)((threads + 255) / 256);
            edge_scatter_kernel<<<blocks, 256, 0, stream>>>(
                (const v4i*)(ei + src_off), (const v4i*)(ei + dst_off),
                s, y, npairs);
        } else {
            int blocks = (int)((E + 255) / 256);
            edge_scatter_scalar_kernel<<<blocks, 256, 0, stream>>>(
                ei + src_off, ei + dst_off, s, y, E);
        }
        if (pass == 0) {
            int blocks = (N + 255) / 256;
            node_mid_kernel<<<blocks, 256, 0, stream>>>(W, s, y, N);
        }
    }

    // K5: sigmoid epilogue
    {
        int blocks = (N + 255) / 256;
        node_out_kernel<<<blocks, 256, 0, stream>>>(fw, y, out, N);
    }
}